// RepulsionFilteredLinear_24172075942269
// MI455X (gfx1250) — compile-verified
//
#include <hip/hip_runtime.h>
#include <hip/hip_bf16.h>
#include <stdint.h>

#define NUM_RBF    64
#define NTYPES     25
#define NPAIR      (NTYPES * NTYPES)        // 625 rows
#define ROW_STRIDE 68                       // 64 + 4 pad: 272B rows stay 16B aligned, kill bank conflicts
#define TBL_FLOATS (NPAIR * ROW_STRIDE)     // 42500 floats in LDS
#define NGRAPHS    64
#define BLOCK      512
#define GRID       512
#define ROWS_TDM   320                      // rows 0..319 staged by the Tensor Data Mover
                                            // rows 320..624 staged by async global->LDS b128

typedef uint32_t u32x4 __attribute__((ext_vector_type(4)));
typedef uint32_t u32x8 __attribute__((ext_vector_type(8)));

// ---------------------------------------------------------------------------
// Kernel 1: fuse rbf_params * radial_filters into one weight table in d_ws,
// and zero the output accumulators (harness poisons d_out with 0xAA).
// ---------------------------------------------------------------------------
__global__ void fuse_tables_kernel(const float* __restrict__ rbf,
                                   const float* __restrict__ filt,
                                   float* __restrict__ wtab, int n,
                                   float* __restrict__ out, int out_n) {
    int i = blockIdx.x * blockDim.x + threadIdx.x;
    if (i < n) wtab[i] = rbf[i] * filt[i];
    if (blockIdx.x == 0 && threadIdx.x < out_n) out[threadIdx.x] = 0.0f;
}

// ---------------------------------------------------------------------------
// Kernel 2: per-edge filtered RBF energy + per-graph segment reduction.
// The 160KB fused table is staged into LDS via BOTH CDNA5 DMA paths:
//   wave 0   -> TDM tensor_load_to_lds (hardware row padding 64->68 DWORDs)
//   waves 1+ -> global_load_async_to_lds_b128
// ---------------------------------------------------------------------------
__global__ __launch_bounds__(BLOCK, 1)
void edge_energy_kernel(const float* __restrict__ pos,
                        const float* __restrict__ wtab,
                        const int*   __restrict__ srcIdx,
                        const int*   __restrict__ dstIdx,
                        const int*   __restrict__ types,
                        const int*   __restrict__ batch,
                        float*       __restrict__ out,
                        int E) {
    extern __shared__ float smem[];
    float* wlds = smem;                 // [NPAIR][ROW_STRIDE]
    float* gacc = smem + TBL_FLOATS;    // [NGRAPHS]

    const int tid = threadIdx.x;
    if (tid < NGRAPHS) gacc[tid] = 0.0f;

    if (tid < 32) {
        // ---- wave 0: one TDM descriptor loads rows [0, ROWS_TDM) with padding ----
        // 1-D tensor of ROWS_TDM*64 f32 elements; TDM inserts 4 DWORDs of LDS
        // padding after every 64 DWORDs -> row stride 68 floats, matching wlds.
        const uint32_t nelem    = (uint32_t)(ROWS_TDM * NUM_RBF);   // 20480
        const uint64_t gaddr    = (uint64_t)(uintptr_t)wtab;
        const uint32_t lds_base = (uint32_t)(uintptr_t)wlds;

        // D# group 0 (ISA 8.3): count=1 | lds_addr | global_addr[56:0] | type=2
        u32x4 g0;
        g0.x = 1u;                                               // count=1, user mode
        g0.y = lds_base;                                         // lds_addr [63:32]
        g0.z = (uint32_t)gaddr;                                  // global_addr[31:0]
        g0.w = (uint32_t)((gaddr >> 32) & 0x01FFFFFFu) | (2u << 30); // addr[56:32], type=2

        // D# group 1 (ISA 8.4)
        u32x8 g1;
        g1.s0 = (2u << 16)        // data_size = 4 bytes
              | (1u << 20)        // pad_enable
              | (5u << 22)        // pad_interval: pad after 64 DWORDs
              | (3u << 25);       // pad_amount: 4 DWORDs (16B)
        g1.s1 = (nelem & 0xFFFFu) << 16;          // tensor_dim0[15:0]  @ bits 63:48
        g1.s2 = ((nelem >> 16) & 0xFFFFu)         // tensor_dim0[31:16] @ bits 79:64
              | (1u << 16);                       // tensor_dim1 = 1    @ bits 111:80 (lo16)
        g1.s3 = (nelem << 16);                    // tile_dim0 = nelem  @ bits 127:112
        g1.s4 = 1u;                               // tile_dim1 = 1; tile_dim2 = 0 (2-D tile)
        g1.s5 = nelem;                            // tensor_dim0_stride[31:0]
        g1.s6 = 0u;
        g1.s7 = 0u;

        asm volatile("tensor_load_to_lds %0, %1" :: "s"(g0), "s"(g1) : "memory");
        __builtin_amdgcn_s_wait_tensorcnt(0);
    } else {
        // ---- waves 1..15: async-stage rows [ROWS_TDM, NPAIR) in b128 chunks ----
        const int first_chunk = ROWS_TDM * 16;          // float4 chunks, 16 per row
        const int chunks      = NPAIR * 16;
        for (int i = first_chunk + (tid - 32); i < chunks; i += (BLOCK - 32)) {
            int pair = i >> 4;
            int j    = i & 15;
            unsigned ldsOff = (unsigned)(uintptr_t)(wlds + pair * ROW_STRIDE + 4 * j);
            const float* gp = wtab + i * 4;             // 16B-aligned global source
            asm volatile("global_load_async_to_lds_b128 %0, %1, off"
                         :: "v"(ldsOff), "v"(gp)
                         : "memory");
        }
        asm volatile("s_wait_asynccnt 0" ::: "memory");
    }
    __syncthreads();

    // ---- constants for windowed Gaussian expansion ----
    // gamma = (64/5)^2, mu_k = k*(5/63).  gamma*(d-mu_k)^2 = (64/63)^2*(u-k)^2,
    // u = d*63/5.  In exp2 space: C2 = (64/63)^2 * log2(e).
    const float USCALE = 12.6f;         // 63/5
    const float C2     = 1.4888668f;    // (64/63)^2 * log2(e)

    // contiguous chunk per block -> coalesced edge_index loads
    int per = (E + (int)gridDim.x - 1) / (int)gridDim.x;
    int e0  = (int)blockIdx.x * per;
    int e1  = e0 + per; if (e1 > E) e1 = E;

    for (int e = e0 + tid; e < e1; e += BLOCK) {
        int s = srcIdx[e];
        int r = dstIdx[e];
        float dx = pos[3 * r + 0] - pos[3 * s + 0];
        float dy = pos[3 * r + 1] - pos[3 * s + 1];
        float dz = pos[3 * r + 2] - pos[3 * s + 2];
        float d  = __builtin_sqrtf(dx * dx + dy * dy + dz * dz + 1e-12f);

        // cosine-cutoff envelope is exactly 0 for d >= 5: skip ~71% of edges
        if (d < 5.0f) {
            int ts = types[s];
            int tr = types[r];
            int lo = ts < tr ? ts : tr;
            int hi = ts < tr ? tr : ts;
            const float* wrow = wlds + (lo * NTYPES + hi) * ROW_STRIDE;

            float u  = d * USCALE;             // in [0, 63)
            int   k0 = (int)u - 5;
            k0 = k0 < 0 ? 0 : (k0 > NUM_RBF - 12 ? NUM_RBF - 12 : k0);
            float t  = u - (float)k0;

            float acc = 0.0f;
#pragma unroll
            for (int j = 0; j < 12; ++j) {
                // exp(-gamma*(d-mu_k)^2) == 2^(-C2*t^2); one v_exp_f32 each
                float v = __builtin_amdgcn_exp2f(-C2 * t * t);
                acc = __builtin_fmaf(v, wrow[k0 + j], acc);
                t -= 1.0f;
            }
            // fc = 0.5*(cos(pi*d/5)+1); __cosf -> v_cos_f32
            float fc = 0.5f * (__cosf(0.62831853f * d) + 1.0f);
            atomicAdd(&gacc[batch[s]], acc * fc);   // ds_add_f32
        }
    }

    __syncthreads();
    if (tid < NGRAPHS) atomicAdd(&out[tid], gacc[tid]);
}

// ---------------------------------------------------------------------------
// Entry point
// ---------------------------------------------------------------------------
extern "C" void kernel_launch(void* const* d_in, const int* in_sizes, int n_in,
                              void* d_out, int out_size, void* d_ws, size_t ws_size,
                              hipStream_t stream) {
    const float* pos   = (const float*)d_in[0];   // [N,3]
    const float* rbf   = (const float*)d_in[1];   // [25,25,64]
    const float* filt  = (const float*)d_in[2];   // [25,25,64]
    const int*   eidx  = (const int*)d_in[3];     // [2,E]
    const int*   types = (const int*)d_in[4];     // [N]
    const int*   batch = (const int*)d_in[5];     // [N]
    float*       out   = (float*)d_out;           // [64]
    float*       wtab  = (float*)d_ws;            // fused table, 160 KB

    const int E = in_sizes[3] / 2;
    const int n = NPAIR * NUM_RBF;

    fuse_tables_kernel<<<(n + 255) / 256, 256, 0, stream>>>(
        rbf, filt, wtab, n, out, out_size);

    const size_t shmem = (size_t)(TBL_FLOATS + NGRAPHS) * sizeof(float);
    edge_energy_kernel<<<GRID, BLOCK, shmem, stream>>>(
        pos, wtab, eidx, eidx + E, types, batch, out, E);
}